// Arm_3D_Dyn_16492674417209
// MI455X (gfx1250) — compile-verified
//
#include <hip/hip_runtime.h>
#include <hip/hip_bf16.h>

#define DT      0.01f
#define LAM_DT  0.02f   // LAM * DT

__global__ void __launch_bounds__(256)
arm3d_dyn_kernel(const float4* __restrict__ theta,
                 const float4* __restrict__ vel,
                 const float4* __restrict__ tau,
                 const float*  __restrict__ pL1,
                 const float*  __restrict__ pL2,
                 const float*  __restrict__ pM1,
                 const float*  __restrict__ pM2,
                 float4* __restrict__ out,   // (N,12) = 3x float4 per row
                 int n)
{
    const float L1 = pL1[0], L2 = pL2[0], M1 = pM1[0], M2 = pM2[0];
    const float L1sq = L1 * L1, L2sq = L2 * L2, L1L2 = L1 * L2;
    const float M2L2 = M2 * L2;
    const float ONE3 = 1.0f / 3.0f, ONE6 = 1.0f / 6.0f, ONE12 = 1.0f / 12.0f;

    const int stride = (int)(blockDim.x * gridDim.x);
    for (int i = (int)(blockIdx.x * blockDim.x + threadIdx.x); i < n; i += stride) {
        // --- prefetch next grid-stride iteration (gfx1250 global_prefetch_b8) ---
        int inext = i + stride;
        if (inext < n) {
            __builtin_prefetch((const void*)&theta[inext], 0, 3);
            __builtin_prefetch((const void*)&vel[inext],   0, 3);
            __builtin_prefetch((const void*)&tau[inext],   0, 3);
        }

        const float4 th = theta[i];   // global_load_b128
        const float4 vl = vel[i];
        const float4 tq = tau[i];

        const float t2 = th.y, t3 = th.z, t4 = th.w;
        const float dt1 = vl.x, dt2 = vl.y, dt3 = vl.z, dt4 = vl.w;

        // |t| <= 1.5: hardware v_sin/v_cos, no range-reduction loss.
        // Error reaches output only via DT*qdd (damped by 1e-2).
        const float s2 = __sinf(t2), c2 = __cosf(t2);
        const float s3 = __sinf(t3), c3 = __cosf(t3);
        const float s4 = __sinf(t4), c4 = __cosf(t4);

        const float s2s2 = s2 * s2, c2c2 = c2 * c2;
        const float s3s3 = s3 * s3, c3c3 = c3 * c3;
        const float s4s4 = s4 * s4, c4c4 = c4 * c4;
        const float s2c2 = s2 * c2, s4c4 = s4 * c4;
        const float c3c4 = c3 * c4, s3c4 = s3 * c4;

        // ---------------- mass matrix D (symmetric, d23 = 0) ----------------
        //  cos(2t2-t4)-cos(2t2+t4)                      = 4 s2 c2 s4
        //  cos(-2t2+t3+2t4)-cos(2t2-t3+2t4)
        //   +cos(2t2+t3-2t4)-cos(2t2+t3+2t4)            = 16 s2 c2 c3 s4 c4
        //  cos(t3-2t4)-cos(t3+2t4)                      = 4 s3 s4 c4
        const float d00 = L1sq * M1 * c2c2 * ONE3
            + M2 * ONE3 * (3.0f * L1sq * c2c2
                           - 3.0f * L1L2 * s2c2 * s4
                           + 3.0f * L1L2 * c2c2 * c3c4
                           - 2.0f * L2sq * s2c2 * c3 * s4c4
                           + L2sq * (c2c2 * c3c3 * c4c4 + c2c2 * c4c4 - c2c2
                                     - c3c3 * c4c4 + 1.0f));
        const float Bf = 3.0f * L1 * s2 + 2.0f * L2 * s2 * c3c4 + 2.0f * L2 * s4 * c2;
        const float d01 = M2L2 * Bf * s3c4 * ONE6;
        const float d02 = M2L2 * (3.0f * L1 * c2 * c3 - 2.0f * L2 * s2 * s4 * c3
                                  + 2.0f * L2 * c2 * c4) * c4 * ONE6;
        const float d03 = M2L2 * (2.0f * L2 * s2 - 3.0f * L1 * s4 * c2) * s3 * ONE6;
        const float d11 = L1sq * M1 * ONE3
            + M2 * ONE3 * (3.0f * L1sq + 3.0f * L1L2 * c3c4
                           + L2sq * (s3s3 * s4s4 - s3s3 + 1.0f));
        const float d12 = L2sq * M2 * s3 * s4c4 * ONE3;
        const float d13 = M2L2 * (3.0f * L1 * c4 + 2.0f * L2 * c3) * ONE6;
        const float d22 = L2sq * M2 * c4c4 * ONE3;
        const float d33 = L2sq * M2 * ONE3;

        // ---------------- bias terms ----------------
        // sin(2t2-t4)+sin(2t2+t4) = 4 s2 c2 c4 ; sin(2t2-t3)+sin(2t2+t3) = 4 s2 c2 c3
        // sin(t3-2t4)-sin(t3+2t4) = -4 c3 s4 c4 ; sin(2t2) = 2 s2 c2 ; sin(2t4) = 2 s4 c4
        const float B14 = 12.0f * L1 * s2c2 * c4 + 12.0f * L1 * s4 * c2c2 * c3
                        - 8.0f * L2 * s2c2 * c3 + 16.0f * L2 * s2c2 * c3 * c4c4
                        + 8.0f * L2 * s4 * c2c2 * c3c3 * c4
                        + 8.0f * L2 * s4 * c2c2 * c4
                        - 8.0f * L2 * s4 * c3c3 * c4;
        const float G = 4.0f * L1sq * M1 * s2c2
            + M2 * (12.0f * L1sq * s2c2
                    + 12.0f * L1L2 * s2c2 * c3c4
                    + 12.0f * L1L2 * s4 * c2c2
                    - 6.0f * L1L2 * s4
                    - 4.0f * L2sq * c3 * s4c4
                    + 4.0f * L2sq * s2c2 * c3c3 * c4c4
                    + 4.0f * L2sq * s2c2 * c4c4
                    - 4.0f * L2sq * s2c2
                    + 8.0f * L2sq * s4 * c2c2 * c3c4);

        const float h0 =
              -0.5f * L1L2 * M2 * dt4 * dt4 * s3 * c2 * c4
            - (2.0f * ONE3) * L2sq * M2 * dt2 * dt3 * s2 * s3s3 * c4c4
            - (2.0f * ONE3) * L2sq * M2 * dt2 * dt4 * (s2 * s4 * c3 - c2 * c4) * s3c4
            + M2L2 * dt1 * dt3 * (6.0f * L1 * s2s2 - 6.0f * L1
                                  + 4.0f * L2 * s2c2 * s4
                                  + 4.0f * L2 * s2s2 * c3c4) * s3c4 * ONE6
            - M2L2 * dt1 * dt4 * B14 * ONE12
            + M2L2 * dt2 * dt2 * (3.0f * L1 * c2 - 2.0f * L2 * s2 * s4
                                  + 2.0f * L2 * c2 * c3c4) * s3c4 * ONE6
            - M2L2 * dt3 * dt3 * (3.0f * L1 * c2 - 2.0f * L2 * s2 * s4) * s3c4 * ONE6
            - M2L2 * dt3 * dt4 * (3.0f * L1 * c2 * c3 - 2.0f * L2 * s2 * s4 * c3
                                  + 2.0f * L2 * c2 * c4) * s4 * ONE3
            - dt1 * dt2 * G * ONE6;

        const float h1 =
              -0.5f * L1L2 * M2 * dt4 * dt4 * s4
            + L2sq * M2 * dt3 * dt3 * c3 * s4c4 * ONE3
            - (2.0f * ONE3) * L2sq * M2 * dt3 * dt4 * s3 * s4s4
            + M2L2 * dt1 * dt3 * Bf * c3c4 * ONE3
            - M2L2 * dt1 * dt4 * Bf * s3 * s4 * ONE3
            - M2L2 * dt2 * dt3 * (3.0f * L1 + 2.0f * L2 * c3c4) * s3c4 * ONE3
            - M2L2 * dt2 * dt4 * (3.0f * L1 * c3 - 2.0f * L2 * s3s3 * c4) * s4 * ONE3
            + dt1 * dt1 * G * ONE12;

        const float h2 = M2L2 * ONE6 * (
              -4.0f * L2 * dt1 * dt4 * (s2 * c3c4 + s4 * c2) * c4
            + 4.0f * L2 * dt2 * dt4 * s3 * c4c4
            - 4.0f * L2 * dt3 * dt4 * s4c4
            - dt1 * dt1 * (3.0f * L1 * s2s2 - 3.0f * L1
                           + 2.0f * L2 * s2c2 * s4
                           + 2.0f * L2 * s2s2 * c3c4) * s3c4
            - 2.0f * dt1 * dt2 * Bf * c3c4
            + dt2 * dt2 * (3.0f * L1 + 2.0f * L2 * c3c4) * s3c4);

        const float h3 = M2L2 * ONE6 * (
              4.0f * L2 * dt1 * dt3 * (s2 * c3c4 + s4 * c2) * c4
            - 4.0f * L2 * dt2 * dt3 * s3 * c4c4
            + 2.0f * L2 * dt3 * dt3 * s4c4
            + 0.25f * dt1 * dt1 * B14
            + 2.0f * dt1 * dt2 * Bf * s3 * s4
            + dt2 * dt2 * (3.0f * L1 * c3 - 2.0f * L2 * s3s3 * c4) * s4);

        // ---------------- solve D qdd = tau - h (SPD, no-pivot elimination) --
        float r0 = tq.x - h0, r1 = tq.y - h1, r2 = tq.z - h2, r3 = tq.w - h3;

        const float i00 = __builtin_amdgcn_rcpf(d00);
        const float m10 = d01 * i00, m20 = d02 * i00, m30 = d03 * i00;
        const float b11 = d11 - m10 * d01;
        const float b12 = d12 - m10 * d02;
        const float b13 = d13 - m10 * d03;
        const float b22 = d22 - m20 * d02;
        const float b23 =      - m20 * d03;      // d23 == 0
        const float b33 = d33 - m30 * d03;
        r1 -= m10 * r0;  r2 -= m20 * r0;  r3 -= m30 * r0;

        const float i11 = __builtin_amdgcn_rcpf(b11);
        const float m21 = b12 * i11, m31 = b13 * i11;
        const float c22n = b22 - m21 * b12;
        const float c23n = b23 - m21 * b13;
        const float c33n = b33 - m31 * b13;
        r2 -= m21 * r1;  r3 -= m31 * r1;

        const float i22 = __builtin_amdgcn_rcpf(c22n);
        const float m32 = c23n * i22;
        const float e33 = c33n - m32 * c23n;
        r3 -= m32 * r2;

        const float x3 = r3 * __builtin_amdgcn_rcpf(e33);
        const float x2 = (r2 - c23n * x3) * i22;
        const float x1 = (r1 - b12 * x2 - b13 * x3) * i11;
        const float x0 = (r0 - d01 * x1 - d02 * x2 - d03 * x3) * i00;

        // ---------------- integrate + write (3x global_store_b128) ----------
        float4 o0, o1, o2;
        o0.x = th.x + DT * vl.x;  o0.y = th.y + DT * vl.y;
        o0.z = th.z + DT * vl.z;  o0.w = th.w + DT * vl.w;
        o1.x = vl.x + DT * x0;    o1.y = vl.y + DT * x1;
        o1.z = vl.z + DT * x2;    o1.w = vl.w + DT * x3;
        o2.x = tq.x - LAM_DT * tq.x;  o2.y = tq.y - LAM_DT * tq.y;
        o2.z = tq.z - LAM_DT * tq.z;  o2.w = tq.w - LAM_DT * tq.w;

        out[3 * i + 0] = o0;
        out[3 * i + 1] = o1;
        out[3 * i + 2] = o2;
    }
}

extern "C" void kernel_launch(void* const* d_in, const int* in_sizes, int n_in,
                              void* d_out, int out_size, void* d_ws, size_t ws_size,
                              hipStream_t stream) {
    (void)n_in; (void)d_ws; (void)ws_size; (void)out_size;
    const float4* theta = (const float4*)d_in[0];
    const float4* vel   = (const float4*)d_in[1];
    const float4* tau   = (const float4*)d_in[2];
    const float*  L1    = (const float*)d_in[3];
    const float*  L2    = (const float*)d_in[4];
    const float*  M1    = (const float*)d_in[5];
    const float*  M2    = (const float*)d_in[6];
    float4* out = (float4*)d_out;

    const int n = in_sizes[0] / 4;          // rows
    const int threads = 256;                // 8 wave32 per block
    int blocks = (n + threads - 1) / threads;
    if (blocks > 2048) blocks = 2048;       // grid-stride; keeps prefetch useful
    arm3d_dyn_kernel<<<blocks, threads, 0, stream>>>(theta, vel, tau,
                                                     L1, L2, M1, M2, out, n);
}